// CRF_63428077027637
// MI455X (gfx1250) — compile-verified
//
#include <hip/hip_runtime.h>
#include <math.h>

// CRF NLL on gfx1250 (MI455X). Forward recursion rewritten as exp-space
// f32 WMMA matmuls: part' = feat + m + log( exp(part - m) @ exp(T) ).
//
// Inputs (setup_inputs order):
//   d_in[0] feats       (B,S,TAG) float32
//   d_in[1] mask        (B,S)     bool (1 byte per element)
//   d_in[2] tags        (B,S)     int32
//   d_in[3] transitions (TAG,TAG) float32
// Output: single float32 scalar (forward_score - gold_score).
// Workspace: expT[64*64] floats, then fPart[64] floats.

typedef __attribute__((ext_vector_type(2))) float v2f;
typedef __attribute__((ext_vector_type(8))) float v8f;

#define B_     64
#define S_     256
#define T_     64
#define START_ 62
#define END_   63
#define PITCH  68   // LDS row pitch (floats) to dodge bank conflicts

// ---------------------------------------------------------------------------
// Kernel 1: expT[i][j] = exp(T[i][j]); NEG entries underflow to exact 0.
// ---------------------------------------------------------------------------
__global__ void crf_expT_kernel(const float* __restrict__ trans,
                                float* __restrict__ expT) {
    int i = blockIdx.x * blockDim.x + threadIdx.x;
    if (i < T_ * T_) expT[i] = expf(trans[i]);
}

// ---------------------------------------------------------------------------
// Kernel 2: forward algorithm. 4 blocks x 128 threads (4 waves).
// Block g handles batches [16g, 16g+16). Wave w owns output columns
// [16w, 16w+16) and chains 16 x V_WMMA_F32_16X16X4_F32 per time step.
// Writes fPart[b] = logsumexp_i(part[b,i] + T[i,END]).
// ---------------------------------------------------------------------------
__global__ __launch_bounds__(128) void crf_forward_kernel(
    const float* __restrict__ feats,
    const unsigned char* __restrict__ mask,
    const float* __restrict__ trans,
    const float* __restrict__ expT,
    float* __restrict__ fPart) {

    __shared__ float p_lds[16][PITCH];   // log-space partition, 16 batches x 64 tags
    __shared__ float eA[16][PITCH];      // exp(part - rowmax)
    __shared__ float mpart[16][8];       // partial row maxima
    __shared__ float mrow[16];           // row maxima
    __shared__ int   lenRow[16];         // sequence lengths

    const int tid  = threadIdx.x;
    const int b0   = blockIdx.x * 16;
    const int lane = tid & 31;
    const int wave = tid >> 5;
    const int l    = lane & 15;          // lane within half-wave
    const int hi   = lane >> 4;          // half-wave select
    const int n0   = wave * 16;          // this wave's N-tile base column

    // --- sequence lengths (mask rows are contiguous prefixes) ---
    if (tid < 16) {
        int len = 0;
        const unsigned char* mr = mask + (size_t)(b0 + tid) * S_;
        for (int t = 0; t < S_; ++t) len += mr[t];
        lenRow[tid] = len;
    }

    // --- init: part[b,j] = feats[b,0,j] + T[START,j] ---
    const int r  = tid >> 3;             // row (batch) this thread maintains
    const int c0 = (tid & 7) * 8;        // 8-column segment
    for (int c = c0; c < c0 + 8; ++c)
        p_lds[r][c] = feats[((size_t)(b0 + r) * S_) * T_ + c] + trans[START_ * T_ + c];

    // --- preload this wave's B fragments: expT[k][n0+..], kept in VGPRs ---
    // Layout (ISA 7.12.2, 32-bit 4x16): VGPR0 = rows K,K+... lanes0-15 hold
    // K=4k (x) / 4k+1 (y); lanes16-31 hold K=4k+2 (x) / 4k+3 (y).
    v2f bfrag[16];
#pragma unroll
    for (int k = 0; k < 16; ++k) {
        const int kr = 4 * k + 2 * hi;
        bfrag[k].x = expT[(size_t)kr * T_ + n0 + l];
        bfrag[k].y = expT[(size_t)(kr + 1) * T_ + n0 + l];
    }

    for (int t = 1; t < S_; ++t) {
        __syncthreads();                              // partition stable

        // prefetch next step's feats segment into cache hierarchy
        if (t + 1 < S_)
            __builtin_prefetch(&feats[((size_t)(b0 + r) * S_ + t + 1) * T_ + c0], 0, 1);

        // --- row max (2-stage reduction) ---
        float mx = p_lds[r][c0];
        for (int c = c0 + 1; c < c0 + 8; ++c) mx = fmaxf(mx, p_lds[r][c]);
        mpart[r][tid & 7] = mx;
        __syncthreads();
        if (tid < 16) {
            float m2 = mpart[tid][0];
            for (int s = 1; s < 8; ++s) m2 = fmaxf(m2, mpart[tid][s]);
            mrow[tid] = m2;
        }
        __syncthreads();

        // --- exp-space A matrix ---
        {
            const float m2 = mrow[r];
            for (int c = c0; c < c0 + 8; ++c) eA[r][c] = expf(p_lds[r][c] - m2);
        }
        __syncthreads();                              // eA ready

        // --- chained f32 WMMA: acc(16x16) += A(16x4) x B(4x16), K = 0..63 ---
        v8f acc = {};
#pragma unroll
        for (int k = 0; k < 16; ++k) {
            v2f a;
            const int kc = 4 * k + 2 * hi;
            a.x = eA[l][kc];
            a.y = eA[l][kc + 1];
            acc = __builtin_amdgcn_wmma_f32_16x16x4_f32(
                false, a, false, bfrag[k], (short)0, acc, false, false);
        }

        // --- masked partition update from D tile ---
        // D layout: VGPR rr -> M=rr (lanes 0-15) / M=rr+8 (lanes 16-31), N=n0+l
#pragma unroll
        for (int rr = 0; rr < 8; ++rr) {
            const int row = rr + 8 * hi;
            const int col = n0 + l;
            if (t < lenRow[row]) {
                const float fv = feats[((size_t)(b0 + row) * S_ + t) * T_ + col];
                p_lds[row][col] = fv + mrow[row] + logf(acc[rr]);
            }
        }
    }

    __syncthreads();
    // --- final: fPart[b] = logsumexp_i(part[b,i] + T[i,END]) ---
    if (tid < 16) {
        float mq = -3.4e38f;
        for (int i = 0; i < T_; ++i)
            mq = fmaxf(mq, p_lds[tid][i] + trans[i * T_ + END_]);
        float sum = 0.f;
        for (int i = 0; i < T_; ++i)
            sum += expf(p_lds[tid][i] + trans[i * T_ + END_] - mq);
        fPart[b0 + tid] = mq + logf(sum);
    }
}

// ---------------------------------------------------------------------------
// Kernel 3: gold score per batch + deterministic fixed-order reduction.
// ---------------------------------------------------------------------------
__global__ __launch_bounds__(64) void crf_finalize_kernel(
    const float* __restrict__ feats,
    const unsigned char* __restrict__ mask,
    const int* __restrict__ tags,
    const float* __restrict__ trans,
    const float* __restrict__ fPart,
    float* __restrict__ out) {

    __shared__ float gold[B_];
    const int b = threadIdx.x;

    float acc = 0.f;
    int prev = START_;
    int lastTag = START_;
    const int* tg = tags + (size_t)b * S_;
    const unsigned char* mk = mask + (size_t)b * S_;
    const float* fb = feats + (size_t)b * S_ * T_;
    for (int t = 0; t < S_; ++t) {
        if (mk[t]) {
            const int cur = tg[t];
            acc += fb[(size_t)t * T_ + cur] + trans[prev * T_ + cur];
            prev = cur;
            lastTag = cur;
        }
    }
    acc += trans[lastTag * T_ + END_];
    gold[b] = acc;
    __syncthreads();

    if (b == 0) {
        float total = 0.f;
        for (int i = 0; i < B_; ++i) total += fPart[i] - gold[i];
        out[0] = total;
    }
}

// ---------------------------------------------------------------------------
extern "C" void kernel_launch(void* const* d_in, const int* in_sizes, int n_in,
                              void* d_out, int out_size, void* d_ws, size_t ws_size,
                              hipStream_t stream) {
    const float*         feats = (const float*)d_in[0];
    const unsigned char* mask  = (const unsigned char*)d_in[1];
    const int*           tags  = (const int*)d_in[2];
    const float*         trans = (const float*)d_in[3];
    float*               out   = (float*)d_out;

    float* expT  = (float*)d_ws;           // 64*64 floats
    float* fPart = expT + T_ * T_;         // 64 floats

    crf_expT_kernel<<<(T_ * T_ + 255) / 256, 256, 0, stream>>>(trans, expT);
    crf_forward_kernel<<<4, 128, 0, stream>>>(feats, mask, trans, expT, fPart);
    crf_finalize_kernel<<<1, 64, 0, stream>>>(feats, mask, tags, trans, fPart, out);
}